// StyleEncoder_27848567947819
// MI455X (gfx1250) — compile-verified
//
#include <hip/hip_runtime.h>
#include <hip/hip_bf16.h>

// ---------------------------------------------------------------------------
// StyleEncoder fused pipeline for MI455X (gfx1250, wave32).
//
// Key insight: the conv input is piecewise-constant over <=20 regions, so the
// 3x3 convs reduce to per-region per-tap response tables U[s][tap][co]
// (computed with V_WMMA_F32_16X16X4_F32) followed by a bandwidth-bound
// gather-and-add pass that writes the 256MB output at HBM rate.
// ---------------------------------------------------------------------------

#define Bn 8
#define Cn 64
#define Hn 128
#define Wn 128
#define HW 16384      // Hn*Wn
#define S20 20        // number of one-hot classes (seg+1 in [0,19])
#define NTAP 9
#define UN 576        // NTAP*Cn columns of the U GEMM
#define USTRIDE 577   // LDS row stride for U (577 mod 64 == 1 -> conflict-free)
#define UZERO (S20*USTRIDE)        // 11540: start of zeroed "region 20" slot
#define ULDS  (UZERO + UN + 4)     // total padded LDS floats for U table

typedef float v2f __attribute__((ext_vector_type(2)));
typedef float v8f __attribute__((ext_vector_type(8)));

// ---------------------------------------------------------------------------
// Kernel 1: region sums.  grid = 2*Bn*Cn blocks; block (d,b,c) owns
// sums[d][b][*][c] exclusively -> plain stores, no global init needed.
// ---------------------------------------------------------------------------
__global__ void pool_sums_k(const float* __restrict__ featA,
                            const float* __restrict__ featB,
                            const int*   __restrict__ segA,
                            const int*   __restrict__ segB,
                            float* __restrict__ sums) {
    const int d  = blockIdx.x >> 9;        // 0:A 1:B
    const int bc = blockIdx.x & 511;
    const int b  = bc >> 6;
    const int c  = bc & 63;
    const float* feat = d ? featB : featA;
    const int*   seg  = d ? segB  : segA;

    __shared__ float ls[S20];
    if (threadIdx.x < S20) ls[threadIdx.x] = 0.f;
    __syncthreads();

    const int fbase = (b * Cn + c) * HW;
    const int sbase = b * HW;
    for (int p = threadIdx.x; p < HW; p += blockDim.x) {
        const int s = seg[sbase + p] + 1;           // 0..19
        atomicAdd(&ls[s], feat[fbase + p]);
    }
    __syncthreads();
    if (threadIdx.x < S20)
        sums[((d * Bn + b) * S20 + threadIdx.x) * Cn + c] = ls[threadIdx.x];
}

// ---------------------------------------------------------------------------
// Kernel 2: region pixel counts.  grid = 2*Bn blocks.
// ---------------------------------------------------------------------------
__global__ void seg_hist_k(const int* __restrict__ segA,
                           const int* __restrict__ segB,
                           float* __restrict__ cnt) {
    const int d = blockIdx.x >> 3;
    const int b = blockIdx.x & 7;
    const int* seg = d ? segB : segA;

    __shared__ float hs[S20];
    if (threadIdx.x < S20) hs[threadIdx.x] = 0.f;
    __syncthreads();

    const int sbase = b * HW;
    for (int p = threadIdx.x; p < HW; p += blockDim.x)
        atomicAdd(&hs[seg[sbase + p] + 1], 1.f);
    __syncthreads();
    if (threadIdx.x < S20)
        cnt[(d * Bn + b) * S20 + threadIdx.x] = hs[threadIdx.x];
}

// ---------------------------------------------------------------------------
// Kernel 3: style tables.  grid = 2*Bn blocks (d,b).
// style[d][b][s][:] = choose ? relu(rp[s] @ W_fc[s] + b_fc[s])
//                            : relu(glob @ W_fc[0] + b_fc[0])
// ---------------------------------------------------------------------------
__global__ void style_k(const float* __restrict__ sums,
                        const float* __restrict__ cnt,
                        const float* __restrict__ W_fc,   // (20,64,64) [s][c][dd]
                        const float* __restrict__ b_fc,   // (20,64)
                        float* __restrict__ style) {
    const int db = blockIdx.x;            // d*Bn + b
    __shared__ float rp[S20 * Cn];
    __shared__ float glob[Cn];
    __shared__ float rs[S20];
    __shared__ float fcg[Cn];

    // phase 1: region means + global mean
    for (int i = threadIdx.x; i < S20 * Cn; i += blockDim.x) {
        const int s = i >> 6;
        const float cn = cnt[db * S20 + s];
        const float sm = sums[db * S20 * Cn + i];
        rp[i] = (cn > 0.f) ? sm / fmaxf(cn, 1.f) : 0.f;
    }
    if (threadIdx.x < Cn) {
        float t = 0.f;
        for (int s = 0; s < S20; ++s) t += sums[db * S20 * Cn + s * Cn + threadIdx.x];
        glob[threadIdx.x] = t * (1.f / (float)HW);
    }
    __syncthreads();

    // phase 2: row sums (choose test) + global-style FC
    if (threadIdx.x < S20) {
        float t = 0.f;
        for (int c = 0; c < Cn; ++c) t += rp[threadIdx.x * Cn + c];
        rs[threadIdx.x] = t;
    }
    if (threadIdx.x >= 64 && threadIdx.x < 128) {
        const int dd = threadIdx.x - 64;
        float acc = b_fc[dd];
        for (int c = 0; c < Cn; ++c) acc += glob[c] * W_fc[c * Cn + dd]; // W_fc[0]
        fcg[dd] = fmaxf(acc, 0.f);
    }
    __syncthreads();

    // phase 3: per-region FC + select
    for (int i = threadIdx.x; i < S20 * Cn; i += blockDim.x) {
        const int s  = i >> 6;
        const int dd = i & 63;
        float acc = b_fc[s * Cn + dd];
        for (int c = 0; c < Cn; ++c)
            acc += rp[s * Cn + c] * W_fc[(s * Cn + c) * Cn + dd];
        const bool choose = (s != 0) && (rs[s] != 0.f);
        style[db * S20 * Cn + i] = choose ? fmaxf(acc, 0.f) : fcg[dd];
    }
}

// ---------------------------------------------------------------------------
// Kernel 4: U = style(20x64, zero-padded to 32) x Wconv-reshaped(64x576)
// via V_WMMA_F32_16X16X4_F32.  grid = 2(d)*2(g)*Bn = 32 blocks, 288 thr
// (9 waves); wave wv owns N-tiles [4*wv, 4*wv+4), both M-tiles.
// Fragment layouts per CDNA5 ISA 7.12.2:
//   A 16x4 f32: VGPR0 = K0 (lanes0-15) | K2 (lanes16-31); VGPR1 = K1 | K3
//   B 4x16 f32: mirrored (row K in lanes, halves K0/K2, K1/K3)
//   D 16x16 f32: VGPR e -> M=e (lanes0-15) / M=8+e (lanes16-31), N=lane%16
// ---------------------------------------------------------------------------
__global__ void u_gemm_k(const float* __restrict__ style,
                         const float* __restrict__ conv_gamma_w, // (64,64,3,3)
                         const float* __restrict__ conv_beta_w,
                         float* __restrict__ U) {
    const int b = blockIdx.x & 7;
    const int g = (blockIdx.x >> 3) & 1;
    const int d = blockIdx.x >> 4;

    __shared__ float As[32 * Cn];
    const int sbase = (d * Bn + b) * S20 * Cn;
    for (int i = threadIdx.x; i < 32 * Cn; i += blockDim.x)
        As[i] = (i < S20 * Cn) ? style[sbase + i] : 0.f;
    __syncthreads();

    const float* Wg = g ? conv_beta_w : conv_gamma_w;   // [co][c][tap], 576/co
    const int lane  = threadIdx.x & 31;
    const int wv    = threadIdx.x >> 5;                  // 0..8
    const int mrow  = lane & 15;
    const int khalf = (lane >> 4) << 1;                  // 0 or 2
    float* Uo = U + (size_t)((d * 2 + g) * Bn + b) * (S20 * UN);

    for (int nt = wv * 4; nt < wv * 4 + 4; ++nt) {       // 36 N-tiles over 9 waves
        const int n   = nt * 16 + (lane & 15);
        const int co  = n & 63;
        const int tap = n >> 6;
        v8f acc0 = {0.f, 0.f, 0.f, 0.f, 0.f, 0.f, 0.f, 0.f};
        v8f acc1 = {0.f, 0.f, 0.f, 0.f, 0.f, 0.f, 0.f, 0.f};
        for (int kk = 0; kk < 16; ++kk) {
            const int c0 = kk * 4 + khalf;
            v2f a0, a1, bf;
            a0.x = As[mrow * Cn + c0];        a0.y = As[mrow * Cn + c0 + 1];
            a1.x = As[(mrow + 16) * Cn + c0]; a1.y = As[(mrow + 16) * Cn + c0 + 1];
            bf.x = Wg[co * UN + c0 * NTAP + tap];
            bf.y = Wg[co * UN + (c0 + 1) * NTAP + tap];
            acc0 = __builtin_amdgcn_wmma_f32_16x16x4_f32(
                       false, a0, false, bf, (short)0, acc0, false, false);
            acc1 = __builtin_amdgcn_wmma_f32_16x16x4_f32(
                       false, a1, false, bf, (short)0, acc1, false, false);
        }
        const int nbase = nt * 16 + (lane & 15);
        const int mofs  = (lane >> 4) * 8;
        #pragma unroll
        for (int e = 0; e < 8; ++e) {
            const int m = e + mofs;                       // 0..15 -> always valid
            Uo[m * UN + nbase] = acc0[e];
            const int m1 = 16 + m;                        // only rows 16..19 real
            if (m1 < S20) Uo[m1 * UN + nbase] = acc1[e];
        }
    }
}

// ---------------------------------------------------------------------------
// Kernel 5: bandwidth-bound output assembly.
// out[kg][b][co][h][w] = relu(bias[co] + sum_{valid taps} U[s(tap)][tap][co])
// grid = 8(kg)*8(b)*16(row tiles) blocks, 256 thr.  U table in LDS with
// stride 577 (== 1 mod 64 banks) so divergent-s gathers are conflict-free;
// OOB taps read a zeroed LDS slot (branch-free inner loop).
// Lanes map to w -> fully coalesced stores.
// ---------------------------------------------------------------------------
__global__ void final_k(const float* __restrict__ U,
                        const int*   __restrict__ segA,
                        const int*   __restrict__ segB,
                        const float* __restrict__ conv_gamma_b,
                        const float* __restrict__ conv_beta_b,
                        float* __restrict__ out) {
    const int kg = blockIdx.x >> 7;            // 0..7 output map
    const int b  = (blockIdx.x >> 4) & 7;
    const int rt = blockIdx.x & 15;            // 8-row tile
    const int k  = kg >> 1;                    // combo 0..3
    const int g  = kg & 1;                     // gamma/beta

    // combo -> (seg source, mat/style source)
    const int segsel = (k == 1 || k == 3) ? 1 : 0;   // A,B,A,B
    const int dsel   = (k == 1 || k == 2) ? 1 : 0;   // A,B,B,A
    const int* seg = segsel ? segB : segA;
    const float* bias = g ? conv_beta_b : conv_gamma_b;

    __shared__ float Ul[ULDS];
    __shared__ float Bl[Cn];

    const size_t ubase = (size_t)((dsel * 2 + g) * Bn + b) * (S20 * UN);
    for (int e = threadIdx.x; e < S20 * UN; e += blockDim.x)
        Ul[e + e / UN] = U[ubase + e];                 // pad: +1 float per 576
    for (int i = threadIdx.x; i < UN; i += blockDim.x)
        Ul[UZERO + i] = 0.f;                           // "region 20" = zeros
    if (threadIdx.x < Cn) Bl[threadIdx.x] = bias[threadIdx.x];
    __syncthreads();

    const int w  = threadIdx.x & 127;
    const int rh = threadIdx.x >> 7;                   // 0/1 row phase
    const int h0 = rt * 8;
    float* outb = out + (size_t)(kg * Bn + b) * (Cn * HW);
    const int sbase = b * HW;

    for (int rr = rh; rr < 8; rr += 2) {
        const int h = h0 + rr;
        int idx9[NTAP];
        int t = 0;
        #pragma unroll
        for (int dy = -1; dy <= 1; ++dy) {
            #pragma unroll
            for (int dx = -1; dx <= 1; ++dx) {
                const int hn = h + dy, wn = w + dx;
                int id;
                if (hn >= 0 && hn < Hn && wn >= 0 && wn < Wn) {
                    const int s = seg[sbase + hn * Wn + wn] + 1;
                    id = s * USTRIDE + t * Cn;
                } else {
                    id = UZERO + t * Cn;
                }
                idx9[t++] = id;
            }
        }
        float* orow = outb + h * Wn + w;
        for (int co = 0; co < Cn; ++co) {
            float acc = Bl[co];
            #pragma unroll
            for (int tt = 0; tt < NTAP; ++tt) acc += Ul[idx9[tt] + co];
            orow[co * HW] = fmaxf(acc, 0.f);
        }
    }
}

// ---------------------------------------------------------------------------
// Launch.  Workspace (floats): sums 20480 | cnt 320 | style 20480 | U 368640
// = 409,920 floats (~1.64 MB).
// ---------------------------------------------------------------------------
extern "C" void kernel_launch(void* const* d_in, const int* in_sizes, int n_in,
                              void* d_out, int out_size, void* d_ws, size_t ws_size,
                              hipStream_t stream) {
    const float* featA = (const float*)d_in[0];
    const float* featB = (const float*)d_in[1];
    const float* W_fc  = (const float*)d_in[2];
    const float* b_fc  = (const float*)d_in[3];
    const float* cgw   = (const float*)d_in[4];
    const float* cgb   = (const float*)d_in[5];
    const float* cbw   = (const float*)d_in[6];
    const float* cbb   = (const float*)d_in[7];
    const int*   segA  = (const int*)d_in[8];
    const int*   segB  = (const int*)d_in[9];
    float* out = (float*)d_out;

    float* ws    = (float*)d_ws;
    float* sums  = ws;                     // 2*8*20*64
    float* cnt   = ws + 20480;             // 2*8*20
    float* style = ws + 20800;             // 2*8*20*64
    float* U     = ws + 41280;             // 2*2*8*20*576

    pool_sums_k<<<dim3(2 * Bn * Cn), dim3(256), 0, stream>>>(featA, featB, segA, segB, sums);
    seg_hist_k <<<dim3(2 * Bn),      dim3(256), 0, stream>>>(segA, segB, cnt);
    style_k    <<<dim3(2 * Bn),      dim3(256), 0, stream>>>(sums, cnt, W_fc, b_fc, style);
    u_gemm_k   <<<dim3(32),          dim3(288), 0, stream>>>(style, cgw, cbw, U);
    final_k    <<<dim3(8 * Bn * 16), dim3(256), 0, stream>>>(U, segA, segB, cgb, cbb, out);
}